// DecoderCell_16724602651350
// MI455X (gfx1250) — compile-verified
//
#include <hip/hip_runtime.h>
#include <math.h>

typedef float v2f __attribute__((ext_vector_type(2)));
typedef float v8f __attribute__((ext_vector_type(8)));

#define D_   128
#define H_   8
#define DH_  16
#define B_   128
#define N_   512
#define C_   4
#define T_   128
#define NEGV (-1.0e9f)

static __device__ __forceinline__ v8f wmma4(v2f a, v2f b, v8f c) {
  // D = A(16x4) * B(4x16) + C(16x16), all f32, wave32
  return __builtin_amdgcn_wmma_f32_16x16x4_f32(false, a, false, b, (short)0, c,
                                               false, false);
}

// ---------------------------------------------------------------------------
// Fused triple projection: {K1,V,K2}[M,128] = X[M,128] * {Wk1,Wv,Wk2}.
// 8 waves per block, one 16-col tile per wave; A fragments loaded ONCE per
// k-step and reused for all three weights (cuts node-embedding HBM reads 3x).
// ---------------------------------------------------------------------------
__global__ __launch_bounds__(256) void proj3(const float* __restrict__ X,
                                             const float* __restrict__ W0,
                                             const float* __restrict__ W1,
                                             const float* __restrict__ W2,
                                             float* __restrict__ O0,
                                             float* __restrict__ O1,
                                             float* __restrict__ O2) {
  const int lane = threadIdx.x & 31;
  const int l15  = lane & 15;
  const int half = lane >> 4;
  const int nt   = threadIdx.x >> 5;          // col tile 0..7
  const size_t row0 = (size_t)blockIdx.x * 16;
  const int col0 = nt * 16;

  v8f c0 = {}, c1 = {}, c2 = {};
  const float* xp  = X + (row0 + l15) * (size_t)128 + half * 2;
  const size_t wof = (size_t)(half * 2) * 128 + col0 + l15;
  const float* w0 = W0 + wof;
  const float* w1 = W1 + wof;
  const float* w2 = W2 + wof;
  for (int k = 0; k < 128; k += 4) {
    v2f a;
    a.x = xp[k];
    a.y = xp[k + 1];
    v2f b0, b1, b2;
    b0.x = w0[(size_t)k * 128];       b0.y = w0[(size_t)(k + 1) * 128];
    b1.x = w1[(size_t)k * 128];       b1.y = w1[(size_t)(k + 1) * 128];
    b2.x = w2[(size_t)k * 128];       b2.y = w2[(size_t)(k + 1) * 128];
    c0 = wmma4(a, b0, c0);
    c1 = wmma4(a, b1, c1);
    c2 = wmma4(a, b2, c2);
  }
  const size_t oof = (row0 + 8 * half) * 128 + col0 + l15;
  float* o0 = O0 + oof;
  float* o1 = O1 + oof;
  float* o2 = O2 + oof;
#pragma unroll
  for (int r = 0; r < 8; ++r) {
    o0[(size_t)r * 128] = c0[r];
    o1[(size_t)r * 128] = c1[r];
    o2[(size_t)r * 128] = c2[r];
  }
}

// ---------------------------------------------------------------------------
// Generic Out[M,128] = X[M,K] * W[K,128].  8 waves per block, one 16-col tile
// per wave; grid.x = M/16 row tiles.
// ---------------------------------------------------------------------------
__global__ __launch_bounds__(256) void gemm128(const float* __restrict__ X,
                                               const float* __restrict__ W,
                                               float* __restrict__ Out, int K) {
  const int lane = threadIdx.x & 31;
  const int l15  = lane & 15;
  const int half = lane >> 4;
  const int nt   = threadIdx.x >> 5;          // col tile 0..7
  const size_t row0 = (size_t)blockIdx.x * 16;
  const int col0 = nt * 16;

  v8f c = {};
  const float* xp = X + (row0 + l15) * (size_t)K + half * 2;
  const float* wp = W + (size_t)(half * 2) * 128 + col0 + l15;
  for (int k = 0; k < K; k += 4) {
    v2f a, b;
    a.x = xp[k];
    a.y = xp[k + 1];
    b.x = wp[(size_t)k * 128];
    b.y = wp[(size_t)(k + 1) * 128];
    c = wmma4(a, b, c);
  }
  float* op = Out + (row0 + 8 * half) * 128 + col0 + l15;
#pragma unroll
  for (int r = 0; r < 8; ++r) op[(size_t)r * 128] = c[r];
}

// ---------------------------------------------------------------------------
// Q1[m,:] = ctx[m,0:128]*Wq[0:128,:] + ctx[m,128]*Wq[128,:] + Qfix[b(m),:]
// m flattens [T,B,C]; b = (m/C) % B.
// ---------------------------------------------------------------------------
__global__ __launch_bounds__(256) void qstep(const float* __restrict__ ctx,
                                             const float* __restrict__ Wq,
                                             const float* __restrict__ Qfix,
                                             float* __restrict__ Q1) {
  const int lane = threadIdx.x & 31;
  const int l15  = lane & 15;
  const int half = lane >> 4;
  const int nt   = threadIdx.x >> 5;
  const size_t row0 = (size_t)blockIdx.x * 16;
  const int col0 = nt * 16;

  v8f c = {};
  const float* xp = ctx + (row0 + l15) * (size_t)129 + half * 2;
  const float* wp = Wq + (size_t)(half * 2) * 128 + col0 + l15;
  for (int k = 0; k < 128; k += 4) {
    v2f a, b;
    a.x = xp[k];
    a.y = xp[k + 1];
    b.x = wp[(size_t)k * 128];
    b.y = wp[(size_t)(k + 1) * 128];
    c = wmma4(a, b, c);
  }
  const int col   = col0 + l15;
  const float wl  = Wq[(size_t)128 * 128 + col];
#pragma unroll
  for (int r = 0; r < 8; ++r) {
    const size_t m = row0 + r + 8 * half;
    const int b    = (int)((m / C_) % B_);
    Q1[m * 128 + col] = c[r] + ctx[m * 129 + 128] * wl + Qfix[(size_t)b * 128 + col];
  }
}

// ---------------------------------------------------------------------------
// Multi-head attention, one wave per (b, h, 16-query tile).
// scores -> LDS, masked softmax, attn x Vh -> Att (concatenated heads).
// ---------------------------------------------------------------------------
__global__ __launch_bounds__(32) void attn(const float* __restrict__ Q1,
                                           const float* __restrict__ K1,
                                           const float* __restrict__ V,
                                           const unsigned char* __restrict__ mask,
                                           float* __restrict__ Att) {
  __shared__ float sS[16][N_ + 4];
  const int lane = threadIdx.x;
  const int l15  = lane & 15;
  const int half = lane >> 4;
  const int mt = blockIdx.x, h = blockIdx.y, b = blockIdx.z;

  const int qa = mt * 16 + l15;                 // this lane's M row
  const size_t rowA = ((size_t)(qa >> 2) * B_ + b) * C_ + (qa & 3);
  const float* ap = Q1 + rowA * 128 + h * DH_ + half * 2;

  // ---- pass 1: scores tile (16 x 512), scaled by 1/sqrt(16) ----
  for (int ntile = 0; ntile < N_ / 16; ++ntile) {
    const int n0 = ntile * 16;
    v8f c = {};
    const float* kp = K1 + ((size_t)b * N_ + n0 + l15) * 128 + h * DH_ + half * 2;
#pragma unroll
    for (int ks = 0; ks < 4; ++ks) {
      v2f a, bb;
      a.x  = ap[ks * 4];
      a.y  = ap[ks * 4 + 1];
      bb.x = kp[ks * 4];
      bb.y = kp[ks * 4 + 1];
      c = wmma4(a, bb, c);
    }
#pragma unroll
    for (int r = 0; r < 8; ++r) sS[r + 8 * half][n0 + l15] = c[r] * 0.25f;
  }
  __syncthreads();

  // ---- pass 2: masked softmax per query row ----
  for (int m = 0; m < 16; ++m) {
    const int q = mt * 16 + m;
    const unsigned char* mrow =
        mask + (((size_t)(q >> 2) * B_ + b) * C_ + (q & 3)) * N_;
    float mx = -3.0e38f;
    for (int i = lane; i < N_; i += 32) {
      float s = sS[m][i];
      if (mrow[i]) s = NEGV;
      sS[m][i] = s;
      mx = fmaxf(mx, s);
    }
#pragma unroll
    for (int off = 16; off; off >>= 1) mx = fmaxf(mx, __shfl_xor(mx, off, 32));
    float sum = 0.0f;
    for (int i = lane; i < N_; i += 32) sum += __expf(sS[m][i] - mx);
#pragma unroll
    for (int off = 16; off; off >>= 1) sum += __shfl_xor(sum, off, 32);
    const float inv = 1.0f / sum;
    for (int i = lane; i < N_; i += 32) sS[m][i] = __expf(sS[m][i] - mx) * inv;
  }
  __syncthreads();

  // ---- pass 3: O = attn(16x512) x Vh(512x16) ----
  v8f o = {};
  for (int ntile = 0; ntile < N_ / 16; ++ntile) {
    const int n0 = ntile * 16;
    const float* vp = V + ((size_t)b * N_ + n0 + 2 * half) * 128 + h * DH_ + l15;
#pragma unroll
    for (int ks = 0; ks < 4; ++ks) {
      v2f a, bb;
      a.x  = sS[l15][n0 + ks * 4 + 2 * half];
      a.y  = sS[l15][n0 + ks * 4 + 2 * half + 1];
      bb.x = vp[(size_t)(ks * 4) * 128];
      bb.y = vp[(size_t)(ks * 4 + 1) * 128];
      o = wmma4(a, bb, o);
    }
  }
#pragma unroll
  for (int r = 0; r < 8; ++r) {
    const int q = mt * 16 + r + 8 * half;
    const size_t row = ((size_t)(q >> 2) * B_ + b) * C_ + (q & 3);
    Att[row * 128 + h * DH_ + l15] = o[r];
  }
}

// ---------------------------------------------------------------------------
// Pointer head: logits = CLIP*tanh(Q2.K2 / sqrt(128)), mask -> NEG,
// log_softmax over (c,n) groups of 2048.  One wave per (b, 16-query tile);
// 16 queries = 4 complete t-groups (4 cars each).
// ---------------------------------------------------------------------------
__global__ __launch_bounds__(32) void pointer_logits(
    const float* __restrict__ Q2, const float* __restrict__ K2,
    const unsigned char* __restrict__ mask, float* __restrict__ out) {
  __shared__ float sL[16][N_ + 4];
  const int lane = threadIdx.x;
  const int l15  = lane & 15;
  const int half = lane >> 4;
  const int mt = blockIdx.x, b = blockIdx.y;

  const int qa = mt * 16 + l15;
  const size_t rowA = ((size_t)(qa >> 2) * B_ + b) * C_ + (qa & 3);
  const float* ap = Q2 + rowA * 128 + half * 2;

  // ---- raw dot products (16 x 512) ----
  for (int ntile = 0; ntile < N_ / 16; ++ntile) {
    const int n0 = ntile * 16;
    v8f c = {};
    const float* kp = K2 + ((size_t)b * N_ + n0 + l15) * 128 + half * 2;
    for (int k = 0; k < 128; k += 4) {
      v2f a, bb;
      a.x  = ap[k];
      a.y  = ap[k + 1];
      bb.x = kp[k];
      bb.y = kp[k + 1];
      c = wmma4(a, bb, c);
    }
#pragma unroll
    for (int r = 0; r < 8; ++r) sL[r + 8 * half][n0 + l15] = c[r];
  }
  __syncthreads();

  // ---- tanh clip + mask ----
  const float isq = 0.088388347648318447f;  // 1/sqrt(128)
  for (int idx = lane; idx < 16 * N_; idx += 32) {
    const int m = idx >> 9, n = idx & (N_ - 1);
    const int q = mt * 16 + m;
    float s = 10.0f * tanhf(sL[m][n] * isq);
    if (mask[(((size_t)(q >> 2) * B_ + b) * C_ + (q & 3)) * N_ + n]) s = NEGV;
    sL[m][n] = s;
  }
  __syncthreads();

  // ---- per-t log_softmax over 4 rows x 512 = 2048 ----
#pragma unroll
  for (int g = 0; g < 4; ++g) {
    const int m0 = 4 * g;
    const int t  = mt * 4 + g;
    float mx = -3.0e38f;
    for (int i = lane; i < 4 * N_; i += 32)
      mx = fmaxf(mx, sL[m0 + (i >> 9)][i & (N_ - 1)]);
#pragma unroll
    for (int off = 16; off; off >>= 1) mx = fmaxf(mx, __shfl_xor(mx, off, 32));
    float sum = 0.0f;
    for (int i = lane; i < 4 * N_; i += 32)
      sum += __expf(sL[m0 + (i >> 9)][i & (N_ - 1)] - mx);
#pragma unroll
    for (int off = 16; off; off >>= 1) sum += __shfl_xor(sum, off, 32);
    const float lse = mx + logf(sum);
    float* orow = out + ((size_t)t * B_ + b) * (C_ * N_);
    for (int i = lane; i < 4 * N_; i += 32)
      orow[i] = sL[m0 + (i >> 9)][i & (N_ - 1)] - lse;
  }
}

// ---------------------------------------------------------------------------
extern "C" void kernel_launch(void* const* d_in, const int* in_sizes, int n_in,
                              void* d_out, int out_size, void* d_ws,
                              size_t ws_size, hipStream_t stream) {
  (void)in_sizes; (void)n_in; (void)out_size; (void)ws_size;
  const float* node  = (const float*)d_in[0];
  const float* graph = (const float*)d_in[1];
  const float* ctx   = (const float*)d_in[2];
  const unsigned char* mask = (const unsigned char*)d_in[3];
  const float* Wk1 = (const float*)d_in[4];
  const float* Wv  = (const float*)d_in[5];
  const float* Wk2 = (const float*)d_in[6];
  const float* Wqf = (const float*)d_in[7];
  const float* Wout = (const float*)d_in[8];
  const float* Wqs = (const float*)d_in[9];
  float* out = (float*)d_out;

  const size_t NODE = (size_t)B_ * N_ * D_;   // 8,388,608
  const size_t QSZ  = (size_t)T_ * B_ * C_ * D_;

  float* ws   = (float*)d_ws;
  float* Qfix = ws;                 // B*D
  float* K1   = Qfix + (size_t)B_ * D_;
  float* V    = K1 + NODE;
  float* K2   = V + NODE;
  float* Q1   = K2 + NODE;          // [T,B,C,D]
  float* Q2   = Q1 + QSZ;
  float* Att  = Q1;                 // alias: each attn block reads then writes
                                    // only its own rows x head-slice

  const int MtN = (B_ * N_) / 16;        // 4096 row tiles
  const int MtQ = (T_ * B_ * C_) / 16;   // 4096 row tiles

  proj3<<<MtN, 256, 0, stream>>>(node, Wk1, Wv, Wk2, K1, V, K2);
  gemm128<<<B_ / 16, 256, 0, stream>>>(graph, Wqf, Qfix, 128);
  qstep<<<MtQ, 256, 0, stream>>>(ctx, Wqs, Qfix, Q1);
  attn<<<dim3((T_ * C_) / 16, H_, B_), 32, 0, stream>>>(Q1, K1, V, mask, Att);
  gemm128<<<MtQ, 256, 0, stream>>>(Att, Wout, Q2, 128);
  pointer_logits<<<dim3((T_ * C_) / 16, B_), 32, 0, stream>>>(Q2, K2, mask, out);
}